// PointNet2FeaturePropagation_8246337208758
// MI455X (gfx1250) — compile-verified
//
#include <hip/hip_runtime.h>
#include <hip/hip_bf16.h>

typedef __attribute__((ext_vector_type(2))) float v2f;
typedef __attribute__((ext_vector_type(8))) float v8f;

// Dimensions fixed by reference setup_inputs()
constexpr int Bd = 4, Sd = 4096, Nd = 16384;
constexpr int Cs = 256, Co = 128, Cin = 384, C1 = 256, C2 = 256;
constexpr int Md = Bd * Nd;              // 65536 rows total
constexpr float BN_EPS = 1e-5f;
constexpr float W_EPS  = 1e-8f;

// ---------------------------------------------------------------------------
// kNN: one thread per query point; keys tiled through LDS (broadcast reads).
// Maintains exact smallest-3 squared distances with stable (lowest-index) ties.
// ---------------------------------------------------------------------------
__global__ __launch_bounds__(256) void knn_kernel(
    const float* __restrict__ skey,   // [B,S,3] sampled_xyz
    const float* __restrict__ qry,    // [B,N,3] original_xyz
    int*   __restrict__ oidx,         // [B*N,3]
    float* __restrict__ ow)           // [B*N,3] normalized weights
{
    __shared__ float kt[256 * 3];
    const int bpb = Nd / 256;
    const int b   = blockIdx.x / bpb;
    const int q   = (blockIdx.x % bpb) * 256 + threadIdx.x;
    const int qg  = b * Nd + q;
    const float qx = qry[qg * 3 + 0];
    const float qy = qry[qg * 3 + 1];
    const float qz = qry[qg * 3 + 2];

    float d0 = 3.4e38f, d1 = 3.4e38f, d2 = 3.4e38f;
    int   i0 = 0, i1 = 0, i2 = 0;

    for (int s0 = 0; s0 < Sd; s0 += 256) {
        __syncthreads();
        const float* src = skey + (size_t)(b * Sd + s0) * 3;
        for (int i = threadIdx.x; i < 768; i += 256) kt[i] = src[i];
        __syncthreads();
        #pragma unroll 4
        for (int j = 0; j < 256; ++j) {
            const float dx = qx - kt[j * 3 + 0];
            const float dy = qy - kt[j * 3 + 1];
            const float dz = qz - kt[j * 3 + 2];
            const float d  = dx * dx + dy * dy + dz * dz;
            const int   s  = s0 + j;
            if (d < d2) {
                if (d < d1) {
                    d2 = d1; i2 = i1;
                    if (d < d0) { d1 = d0; i1 = i0; d0 = d; i0 = s; }
                    else        { d1 = d;  i1 = s; }
                } else { d2 = d; i2 = s; }
            }
        }
    }
    const float w0 = 1.f / (d0 + W_EPS);
    const float w1 = 1.f / (d1 + W_EPS);
    const float w2 = 1.f / (d2 + W_EPS);
    const float inv = 1.f / (w0 + w1 + w2);
    oidx[qg * 3 + 0] = i0; oidx[qg * 3 + 1] = i1; oidx[qg * 3 + 2] = i2;
    ow[qg * 3 + 0] = w0 * inv; ow[qg * 3 + 1] = w1 * inv; ow[qg * 3 + 2] = w2 * inv;
}

// ---------------------------------------------------------------------------
// Interpolate + concat: x0[p] = [original_features(128) | interp(256)]
// ---------------------------------------------------------------------------
__global__ __launch_bounds__(128) void interp_concat_kernel(
    const float* __restrict__ sfeat,  // [B,S,Cs]
    const float* __restrict__ ofeat,  // [B,N,Co]
    const int*   __restrict__ kidx,
    const float* __restrict__ kw,
    float* __restrict__ x0)           // [M,Cin]
{
    const int p = blockIdx.x;         // 0..M-1
    const int b = p / Nd;
    const int t = threadIdx.x;        // 128 threads
    if (t < Co) x0[(size_t)p * Cin + t] = ofeat[(size_t)p * Co + t];
    const int   i0 = kidx[p * 3 + 0], i1 = kidx[p * 3 + 1], i2 = kidx[p * 3 + 2];
    const float w0 = kw[p * 3 + 0],  w1 = kw[p * 3 + 1],  w2 = kw[p * 3 + 2];
    const float* f0 = sfeat + ((size_t)b * Sd + i0) * Cs;
    const float* f1 = sfeat + ((size_t)b * Sd + i1) * Cs;
    const float* f2 = sfeat + ((size_t)b * Sd + i2) * Cs;
    for (int c = t; c < Cs; c += 128)
        x0[(size_t)p * Cin + Co + c] = w0 * f0[c] + w1 * f1[c] + w2 * f2[c];
}

// ---------------------------------------------------------------------------
// WMMA fp32 GEMM: Y[M,Nout] = act(A) @ W^T + bias, W row-major [Nout,K].
// Block = 256 threads = 8 waves, tile 64(M) x 64(N), K chunked by 32.
// Waves arranged 4(M) x 2(N); each wave owns TWO 16x16 C tiles (columns
// waveN*32 and waveN*32+16) so each A fragment feeds two
// v_wmma_f32_16x16x4_f32 ops (halves A-side LDS traffic per FLOP).
// FUSED: apply previous layer's BN(scale,shift)+ReLU per K-channel on A-load.
// ---------------------------------------------------------------------------
template <bool FUSED>
__global__ __launch_bounds__(256) void gemm_wmma_kernel(
    const float* __restrict__ A, const float* __restrict__ W,
    const float* __restrict__ bias,
    const float* __restrict__ scale, const float* __restrict__ shift,
    float* __restrict__ Y, int M, int K, int Nout)
{
    // Pads: As row stride 36 floats (144B -> float4-aligned rows; 36r mod 64
    // banks distinct for r=0..15). Bs row stride 65 -> transposed writes hit
    // banks (k+n) mod 64, conflict-free.
    __shared__ float As[64][36];
    __shared__ float Bs[32][65];
    const int tid   = threadIdx.x;
    const int lane  = tid & 31;
    const int wv    = tid >> 5;
    const int waveM = wv & 3;          // 4 tiles along M (rows 16*waveM..+15)
    const int waveN = wv >> 2;         // 2 super-tiles along N (32 cols each)
    const int rowBase = blockIdx.x * 64;
    const int colBase = blockIdx.y * 64;
    const int half = lane >> 4;        // 0: lanes 0-15, 1: lanes 16-31
    const int l15  = lane & 15;
    const int mrow = waveM * 16 + l15;
    const int nc0  = waveN * 32 + l15;     // first 16x16 tile column
    const int nc1  = nc0 + 16;             // second tile column

    v8f acc0 = {};
    v8f acc1 = {};
    for (int kBase = 0; kBase < K; kBase += 32) {
        // Stage A: 64 rows x 32 K-cols as float4 (512 vec4, 2 per thread)
        #pragma unroll
        for (int i = 0; i < 2; ++i) {
            const int linear = i * 256 + tid;
            const int r = linear >> 3, c4 = (linear & 7) * 4;
            const float4 v4 = *(const float4*)&A[(size_t)(rowBase + r) * K + kBase + c4];
            float e0 = v4.x, e1 = v4.y, e2 = v4.z, e3 = v4.w;
            if constexpr (FUSED) {     // fused BN+ReLU of previous layer
                const int kc = kBase + c4;
                e0 = e0 * scale[kc + 0] + shift[kc + 0]; e0 = e0 > 0.f ? e0 : 0.f;
                e1 = e1 * scale[kc + 1] + shift[kc + 1]; e1 = e1 > 0.f ? e1 : 0.f;
                e2 = e2 * scale[kc + 2] + shift[kc + 2]; e2 = e2 > 0.f ? e2 : 0.f;
                e3 = e3 * scale[kc + 3] + shift[kc + 3]; e3 = e3 > 0.f ? e3 : 0.f;
            }
            float4 s4 = make_float4(e0, e1, e2, e3);
            *(float4*)&As[r][c4] = s4;
        }
        // Stage B: Bs[k][n] = W[colBase+n][kBase+k]; 32k x 64n, 8 per thread,
        // contiguous-in-k global reads, transposed LDS writes.
        #pragma unroll
        for (int i = 0; i < 8; ++i) {
            const int linear = i * 256 + tid;
            const int n = linear >> 5, c = linear & 31;
            Bs[c][n] = W[(size_t)(colBase + n) * K + kBase + c];
        }
        __syncthreads();
        #pragma unroll
        for (int kk = 0; kk < 32; kk += 4) {
            v2f a, b0, b1;
            // A 16x4 layout: lane holds (m=l%16, k = 2*half + {0,1})
            a.x = As[mrow][kk + 2 * half];
            a.y = As[mrow][kk + 2 * half + 1];
            // B 4x16 layout: vgpr v, lane l -> (k = v + 2*half, n = l%16)
            b0.x = Bs[kk + 2 * half][nc0];
            b0.y = Bs[kk + 2 * half + 1][nc0];
            b1.x = Bs[kk + 2 * half][nc1];
            b1.y = Bs[kk + 2 * half + 1][nc1];
            acc0 = __builtin_amdgcn_wmma_f32_16x16x4_f32(
                false, a, false, b0, (short)0, acc0, false, false);
            acc1 = __builtin_amdgcn_wmma_f32_16x16x4_f32(
                false, a, false, b1, (short)0, acc1, false, false);
        }
        __syncthreads();
    }
    // C/D layout: vgpr v, lane l -> (row = v + 8*half, col = l%16)
    #pragma unroll
    for (int v = 0; v < 8; ++v) {
        const int r  = rowBase + waveM * 16 + v + 8 * half;
        const int c0 = colBase + nc0;
        const int c1 = colBase + nc1;
        Y[(size_t)r * Nout + c0] = acc0[v] + bias[c0];
        Y[(size_t)r * Nout + c1] = acc1[v] + bias[c1];
    }
}

// ---------------------------------------------------------------------------
// Deterministic per-channel stats: each block writes partial sum/sumsq for a
// fixed 256-row slab (no float atomics -> bitwise-stable across replays).
// ---------------------------------------------------------------------------
__global__ __launch_bounds__(256) void colstats_kernel(
    const float* __restrict__ Y, float* __restrict__ partial, int C)
{
    const int c  = threadIdx.x;          // channel (C == 256)
    const int r0 = blockIdx.x * 256;
    float s = 0.f, sq = 0.f;
    for (int r = 0; r < 256; ++r) {
        const float v = Y[(size_t)(r0 + r) * C + c];
        s += v; sq += v * v;
    }
    partial[blockIdx.x * 512 + c]       = s;
    partial[blockIdx.x * 512 + 256 + c] = sq;
}

__global__ __launch_bounds__(256) void finalize_kernel(
    const float* __restrict__ partial, const float* __restrict__ g,
    const float* __restrict__ beta, float* __restrict__ sc, int M)
{
    const int c = threadIdx.x;
    float s = 0.f, sq = 0.f;
    for (int blk = 0; blk < 256; ++blk) {
        s  += partial[blk * 512 + c];
        sq += partial[blk * 512 + 256 + c];
    }
    const float mean = s / (float)M;
    const float var  = sq / (float)M - mean * mean;   // biased (torch default)
    const float scv  = g[c] * rsqrtf(var + BN_EPS);
    sc[c]       = scv;
    sc[256 + c] = beta[c] - mean * scv;
}

__global__ __launch_bounds__(256) void bn_relu_inplace_kernel(
    float* __restrict__ Y, const float* __restrict__ sc, int C)
{
    const int c = threadIdx.x;
    const size_t i = (size_t)blockIdx.x * C + c;
    const float v = Y[i] * sc[c] + sc[256 + c];
    Y[i] = v > 0.f ? v : 0.f;
}

// ---------------------------------------------------------------------------
extern "C" void kernel_launch(void* const* d_in, const int* in_sizes, int n_in,
                              void* d_out, int out_size, void* d_ws, size_t ws_size,
                              hipStream_t stream)
{
    const float* sampled_xyz      = (const float*)d_in[0];
    const float* sampled_features = (const float*)d_in[1];
    const float* original_xyz     = (const float*)d_in[2];
    const float* original_features= (const float*)d_in[3];
    const float* w0  = (const float*)d_in[4];
    const float* b0  = (const float*)d_in[5];
    const float* g0  = (const float*)d_in[6];
    const float* bt0 = (const float*)d_in[7];
    const float* w1  = (const float*)d_in[8];
    const float* b1  = (const float*)d_in[9];
    const float* g1  = (const float*)d_in[10];
    const float* bt1 = (const float*)d_in[11];
    // d_in[12] = k (always 3 in the reference; hardcoded)

    // Workspace layout (float offsets)
    float* ws      = (float*)d_ws;
    float* x0      = ws;                       // M*Cin            = 25165824
    float* y0      = ws + 25165824;            // M*C1             = 16777216
    int*   kidx    = (int*)(ws + 41943040);    // M*3              =   196608
    float* kw      = ws + 42139648;            // M*3              =   196608
    float* partial = ws + 42336256;            // 256*512          =   131072
    float* sc0     = ws + 42467328;            // scale0|shift0    =      512
    float* sc1     = ws + 42467840;            // scale1|shift1    =      512
    float* out     = (float*)d_out;            // [M, C2]

    // 1) kNN (idx + normalized inverse-distance weights)
    knn_kernel<<<Bd * (Nd / 256), 256, 0, stream>>>(sampled_xyz, original_xyz, kidx, kw);
    // 2) interpolate + concat -> x0 [M, 384]
    interp_concat_kernel<<<Md, 128, 0, stream>>>(sampled_features, original_features,
                                                 kidx, kw, x0);
    // 3) layer0 GEMM: y0 = x0 @ w0^T + b0
    gemm_wmma_kernel<false><<<dim3(Md / 64, C1 / 64), 256, 0, stream>>>(
        x0, w0, b0, nullptr, nullptr, y0, Md, Cin, C1);
    // 4) BN0 stats -> scale0/shift0
    colstats_kernel<<<Md / 256, 256, 0, stream>>>(y0, partial, C1);
    finalize_kernel<<<1, 256, 0, stream>>>(partial, g0, bt0, sc0, Md);
    // 5) layer1 GEMM with fused BN0+ReLU on A-load: out = relu(bn0(y0)) @ w1^T + b1
    gemm_wmma_kernel<true><<<dim3(Md / 64, C2 / 64), 256, 0, stream>>>(
        y0, w1, b1, sc0, sc0 + 256, out, Md, C1, C2);
    // 6) BN1 stats -> scale1/shift1, then apply BN1+ReLU in-place on d_out
    colstats_kernel<<<Md / 256, 256, 0, stream>>>(out, partial, C2);
    finalize_kernel<<<1, 256, 0, stream>>>(partial, g1, bt1, sc1, Md);
    bn_relu_inplace_kernel<<<Md, 256, 0, stream>>>(out, sc1, C2);
}